// PathRaster2d_36610301231400
// MI455X (gfx1250) — compile-verified
//
#include <hip/hip_runtime.h>
#include <hip/hip_bf16.h>

typedef float v2f __attribute__((ext_vector_type(2)));
typedef float v8f __attribute__((ext_vector_type(8)));

#define CANVAS_H 2048
#define CANVAS_W 2048
#define BLOCKS   2048
#define THREADS  256
#define NWAVES   (BLOCKS * THREADS / 32)   // 16384 wave32 waves
// stride per iteration = NWAVES*32 px = 524288 px = exactly 256 rows of 2048

// xor-16 lane swap via v_permlanex16_b32 (identity selects): pure VALU,
// replaces ds_bpermute + s_wait_dscnt.
__device__ __forceinline__ float minf_xor16(float v) {
  int vi = __builtin_bit_cast(int, v);
  int sw = __builtin_amdgcn_permlanex16(vi, vi, 0x76543210, 0xFEDCBA98, false, false);
  return fminf(v, __builtin_bit_cast(float, sw));
}

// D = A(16x4 samples x features) * B(4x16 features x pixels), K=4 rank-3 trick:
// A row s = [-2*sy, -2*sx, sy^2+sx^2, 0]; B col p = [py, px, 1, 0]^T
// D[s][p] + (py^2+px^2) = squared distance pixel p -> sample s.
// D layout: lane l, VGPR j -> sample = j + 8*(l>=16), pixel = l&15.
__global__ __launch_bounds__(THREADS) void PathRaster2d_wmma_kernel(
    const float* __restrict__ kp, float* __restrict__ out) {
  const int lane = (int)(threadIdx.x & 31u);
  const int wave = (int)((blockIdx.x * blockDim.x + threadIdx.x) >> 5);
  const bool hi = lane >= 16;  // lanes 16..31 supply the K=2,3 planes
  const int l15 = lane & 15;

  // Control points scaled to canvas coords (kp is [3][2] = (y,x) normalized).
  const float p0y = kp[0] * (float)CANVAS_H, p0x = kp[1] * (float)CANVAS_W;
  const float p1y = kp[2] * (float)CANVAS_H, p1x = kp[3] * (float)CANVAS_W;
  const float p2y = kp[4] * (float)CANVAS_H, p2x = kp[5] * (float)CANVAS_W;

  // Center on curve centroid to limit cancellation in the expanded |P-S|^2.
  const float cy = (p0y + p1y + p2y) * (1.0f / 3.0f);
  const float cx = (p0x + p1x + p2x) * (1.0f / 3.0f);

  // A operands: two groups of 16 Bezier samples (t = s/31).
  v2f a[2];
#pragma unroll
  for (int g = 0; g < 2; ++g) {
    const float t = (float)(l15 + 16 * g) * (1.0f / 31.0f);
    const float u = 1.0f - t;
    const float b0 = u * u, b1 = 2.0f * t * u, b2 = t * t;
    const float sy = b0 * p0y + b1 * p1y + b2 * p2y - cy;
    const float sx = b0 * p0x + b1 * p1x + b2 * p2x - cx;
    a[g].x = hi ? (sy * sy + sx * sx) : (-2.0f * sy);  // K0 | K2
    a[g].y = hi ? 0.0f : (-2.0f * sx);                 // K1 | K3
  }

  // Spatial decomposition: wave covers 32 consecutive pixels of one row,
  // then strides down 256 rows per iteration (8 iterations total).
  const int base = wave * 32;
  const int x0 = base & (CANVAS_W - 1);
  const int y0 = base >> 11;

  // Loop-invariant per-lane quantities.
  const float px0 = (float)(x0 + l15) - cx;          // tile0 columns
  const float px1 = px0 + 16.0f;                     // tile1 columns
  const float qx  = (float)(x0 + lane) - cx;         // my output column
  const float qx2 = qx * qx;
  v2f b0v, b1v;
  b0v.y = hi ? 0.0f : px0;                           // K1 | K3 (invariant)
  b1v.y = hi ? 0.0f : px1;
  const float maxd_inv = 1.0f / 2896.309375741f;     // 1/sqrt(H^2+W^2)
  const v8f cz = {0.f, 0.f, 0.f, 0.f, 0.f, 0.f, 0.f, 0.f};

  float* orow = out + x0 + lane;
  for (int y = y0; y < CANVAS_H; y += 256) {
    const float yf = (float)y - cy;
    const float bx = hi ? 1.0f : yf;                 // K0 | K2 (only varying piece)
    b0v.x = bx;
    b1v.x = bx;

    v8f d0 = __builtin_amdgcn_wmma_f32_16x16x4_f32(false, a[0], false, b0v,
                                                   (short)0, cz, false, false);
    v8f d1 = __builtin_amdgcn_wmma_f32_16x16x4_f32(false, a[1], false, b0v,
                                                   (short)0, cz, false, false);
    v8f e0 = __builtin_amdgcn_wmma_f32_16x16x4_f32(false, a[0], false, b1v,
                                                   (short)0, cz, false, false);
    v8f e1 = __builtin_amdgcn_wmma_f32_16x16x4_f32(false, a[1], false, b1v,
                                                   (short)0, cz, false, false);

    float m0 = fminf(d0[0], d1[0]);
    float m1 = fminf(e0[0], e1[0]);
#pragma unroll
    for (int i = 1; i < 8; ++i) {
      m0 = fminf(m0, fminf(d0[i], d1[i]));
      m1 = fminf(m1, fminf(e0[i], e1[i]));
    }
    m0 = minf_xor16(m0);  // partner lane holds the other 16 sample rows
    m1 = minf_xor16(m1);

    // Lane l writes pixel (y, x0+l): tile0 for lanes 0-15, tile1 for 16-31.
    const float mu = hi ? m1 : m0;
    const float cc = fmaf(yf, yf, qx2);              // |P|^2 (centered)
    const float d2 = fmaxf(mu + cc, 0.0f);
    const float dist = __builtin_amdgcn_sqrtf(d2);   // raw v_sqrt_f32
    const float val = fmaf(dist, -maxd_inv, 1.0f);
    orow[(size_t)y * CANVAS_W] = (dist < 2.0f) ? val : 0.0f;
  }
}

extern "C" void kernel_launch(void* const* d_in, const int* in_sizes, int n_in,
                              void* d_out, int out_size, void* d_ws, size_t ws_size,
                              hipStream_t stream) {
  (void)in_sizes; (void)n_in; (void)out_size; (void)d_ws; (void)ws_size;
  const float* kp = (const float*)d_in[0];
  float* out = (float*)d_out;
  hipLaunchKernelGGL(PathRaster2d_wmma_kernel, dim3(BLOCKS), dim3(THREADS), 0,
                     stream, kp, out);
}